// DynamicJointOrthogonalityLoss_20169166422878
// MI455X (gfx1250) — compile-verified
//
#include <hip/hip_runtime.h>
#include <hip/hip_bf16.h>
#include <math.h>

typedef __attribute__((ext_vector_type(2))) float v2f;
typedef __attribute__((ext_vector_type(8))) float v8f;

#define M_DIM   64
#define N_DIM   1024
#define W_DIM   512
#define EPSF    1e-12f

// -------------------------------------------------------------------------
// Kernel 1: per-measurement inverse squared norm  s[i,n] = 1/max(||psi_in||,eps)^2
// one wave (32 lanes) per 512-float vector; 8 waves per block.
// -------------------------------------------------------------------------
__global__ __launch_bounds__(256) void k_psi_invnorm2(const float* __restrict__ psi,
                                                      float* __restrict__ s) {
    int vec  = (blockIdx.x * blockDim.x + threadIdx.x) >> 5;   // 0 .. 65535
    int lane = threadIdx.x & 31;
    const float* v = psi + (size_t)vec * W_DIM;
    float ss = 0.f;
    #pragma unroll
    for (int t = 0; t < W_DIM / 32; ++t) {
        float x = v[lane + t * 32];
        ss += x * x;
    }
    #pragma unroll
    for (int o = 16; o; o >>= 1) ss += __shfl_xor(ss, o, 32);
    if (lane == 0) {
        float n = fmaxf(sqrtf(ss), EPSF);
        float r = 1.0f / n;
        s[vec] = r * r;
    }
}

// -------------------------------------------------------------------------
// Kernel 2: g[i,j] = ( <phi_i, phi_j> / (max(|phi_i|,eps) max(|phi_j|,eps)) )^2
// one wave per (i,j) pair.
// -------------------------------------------------------------------------
__global__ __launch_bounds__(32) void k_gram(const float* __restrict__ phi,
                                             float* __restrict__ g) {
    int i = blockIdx.x, j = blockIdx.y, lane = threadIdx.x;
    float sii = 0.f, sjj = 0.f, sij = 0.f;
    #pragma unroll
    for (int t = 0; t < W_DIM / 32; ++t) {
        float a = phi[i * W_DIM + lane + t * 32];
        float b = phi[j * W_DIM + lane + t * 32];
        sii += a * a; sjj += b * b; sij += a * b;
    }
    #pragma unroll
    for (int o = 16; o; o >>= 1) {
        sii += __shfl_xor(sii, o, 32);
        sjj += __shfl_xor(sjj, o, 32);
        sij += __shfl_xor(sij, o, 32);
    }
    if (lane == 0) {
        float ni = fmaxf(sqrtf(sii), EPSF);
        float nj = fmaxf(sqrtf(sjj), EPSF);
        float c  = sij / (ni * nj);
        g[i * M_DIM + j] = c * c;
    }
}

// -------------------------------------------------------------------------
// Kernel 3: initialize loss accumulator to -m*n
// -------------------------------------------------------------------------
__global__ void k_init(float* out) { out[0] = -(float)(M_DIM * N_DIM); }

// -------------------------------------------------------------------------
// Kernel 4: S_i[w,v] = sum_k s[i,k] * psi[i,k,w] * psi[i,k,v]
// GEMM M=N=512, K=1024 per i via V_WMMA_F32_16X16X4_F32, exploiting S_i
// symmetry: only the 10 upper-triangle 128x128 blocks are computed; the
// 6 off-diagonal blocks are mirrored into the lower triangle at store time.
// Block: 256 thr = 8 waves (4M x 2N); wave tile 32x64 (2x4 accumulators);
// block tile 128x128. Grid: 64 i * 10 block pairs = 640 blocks.
// f32 A operand layout: lanes 0-15 hold M=lane (K=0,1 in the 2 VGPRs),
// lanes 16-31 hold M=lane-16 with K=2,3. B mirrors it with lane -> N.
// -------------------------------------------------------------------------
__global__ __launch_bounds__(256) void k_S(const float* __restrict__ psi,
                                           const float* __restrict__ s,
                                           float* __restrict__ Sout) {
    static const int mt_tab[10] = {0,0,0,0,1,1,1,2,2,3};
    static const int nt_tab[10] = {0,1,2,3,1,2,3,2,3,3};

    int bx = blockIdx.x;
    int i  = bx / 10;
    int p  = bx - i * 10;
    int mt = mt_tab[p];          // 128-wide M block
    int nt = nt_tab[p];          // 128-wide N block (nt >= mt)

    int tid    = threadIdx.x;
    int wv     = tid >> 5;
    int lane   = tid & 31;
    int lane16 = lane & 15;
    int khalf  = (lane >> 4) << 1;   // 0 or 2

    int m_base = mt * 128 + (wv >> 1) * 32;   // wave covers 32 M rows
    int n_base = nt * 128 + (wv & 1) * 64;    // wave covers 64 N cols

    const float* psi_i = psi + (size_t)i * N_DIM * W_DIM;
    const float* s_i   = s + i * N_DIM;

    v8f c[2][4] = {};

    for (int k = 0; k < N_DIM; k += 4) {
        const float* r0 = psi_i + (size_t)(k + khalf) * W_DIM;
        const float* r1 = r0 + W_DIM;
        float s0 = s_i[k + khalf];
        float s1 = s_i[k + khalf + 1];

        v2f a[2], b[4];
        #pragma unroll
        for (int tm = 0; tm < 2; ++tm) {
            int m = m_base + tm * 16 + lane16;
            a[tm][0] = r0[m] * s0;
            a[tm][1] = r1[m] * s1;
        }
        #pragma unroll
        for (int tn = 0; tn < 4; ++tn) {
            int v = n_base + tn * 16 + lane16;
            b[tn][0] = r0[v];
            b[tn][1] = r1[v];
        }
        #pragma unroll
        for (int tm = 0; tm < 2; ++tm)
            #pragma unroll
            for (int tn = 0; tn < 4; ++tn)
                c[tm][tn] = __builtin_amdgcn_wmma_f32_16x16x4_f32(
                    false, a[tm], false, b[tn], (short)0, c[tm][tn], false, false);
    }

    // C layout: VGPR r, lanes 0-15 -> (M=r, N=lane); lanes 16-31 -> (M=r+8, N=lane-16)
    float* Si = Sout + ((size_t)i << 18);   // i * 512*512
    int rofs = (lane >> 4) << 3;

    // direct store (row-coalesced across lanes)
    #pragma unroll
    for (int tm = 0; tm < 2; ++tm)
        #pragma unroll
        for (int tn = 0; tn < 4; ++tn)
            #pragma unroll
            for (int r = 0; r < 8; ++r) {
                int Mrow = m_base + tm * 16 + r + rofs;
                int Ncol = n_base + tn * 16 + lane16;
                Si[(size_t)Mrow * W_DIM + Ncol] = c[tm][tn][r];
            }

    // mirrored store into the lower triangle: per lane the 8 accumulator
    // values target 8 consecutive floats (rows are contiguous after the
    // transpose), so emit two b128 stores; lanes L and L+16 complete the
    // 64B line.
    if (mt != nt) {
        #pragma unroll
        for (int tm = 0; tm < 2; ++tm)
            #pragma unroll
            for (int tn = 0; tn < 4; ++tn) {
                int Ncol  = n_base + tn * 16 + lane16;
                int Mbase = m_base + tm * 16 + rofs;
                float* p0 = Si + (size_t)Ncol * W_DIM + Mbase;
                float4 lo = make_float4(c[tm][tn][0], c[tm][tn][1],
                                        c[tm][tn][2], c[tm][tn][3]);
                float4 hi = make_float4(c[tm][tn][4], c[tm][tn][5],
                                        c[tm][tn][6], c[tm][tn][7]);
                *(float4*)(p0)     = lo;
                *(float4*)(p0 + 4) = hi;
            }
    }
}

// -------------------------------------------------------------------------
// Kernel 5: b = Sflat * Sflat^T  (64 x 262144), fused with g-weighting and
// scalar reduction. A and B operand layouts are mirrored (lane->M vs
// lane->N, VGPRs->K), so one fetched fragment serves as both operands.
// b and g are symmetric, so each wave computes only the 10 tiles with
// tm >= tn and doubles the off-diagonal contribution in the reduction.
// Grid: 256 blocks * 8 waves; each wave covers 128 K (32 WMMA steps).
// -------------------------------------------------------------------------
__global__ __launch_bounds__(256) void k_b(const float* __restrict__ Sflat,
                                           const float* __restrict__ g,
                                           float* __restrict__ out) {
    const size_t K = (size_t)W_DIM * W_DIM;  // 262144
    int tid    = threadIdx.x;
    int wv     = tid >> 5;
    int lane   = tid & 31;
    int lane16 = lane & 15;
    int khalf  = (lane >> 4) << 1;

    size_t kbase = (size_t)blockIdx.x * 1024 + (size_t)wv * 128;

    v8f c[4][4] = {};

    for (int kk = 0; kk < 128; kk += 4) {
        size_t k = kbase + kk + khalf;
        v2f f[4];
        #pragma unroll
        for (int t = 0; t < 4; ++t) {
            const float* row = Sflat + (size_t)(t * 16 + lane16) * K;
            f[t][0] = row[k];
            f[t][1] = row[k + 1];
        }
        #pragma unroll
        for (int tm = 0; tm < 4; ++tm)
            #pragma unroll
            for (int tn = 0; tn <= tm; ++tn)
                c[tm][tn] = __builtin_amdgcn_wmma_f32_16x16x4_f32(
                    false, f[tm], false, f[tn], (short)0, c[tm][tn], false, false);
    }

    // weighted partial sum: loss_part = sum_ij g[i,j] * b_part[i,j]
    // (off-diagonal tiles counted twice by symmetry of b and g)
    float part = 0.f;
    int rofs = (lane >> 4) << 3;
    #pragma unroll
    for (int tm = 0; tm < 4; ++tm)
        #pragma unroll
        for (int tn = 0; tn <= tm; ++tn) {
            float wgt = (tm == tn) ? 1.0f : 2.0f;
            #pragma unroll
            for (int r = 0; r < 8; ++r) {
                int ii = tm * 16 + r + rofs;
                int jj = tn * 16 + lane16;
                part += wgt * g[ii * M_DIM + jj] * c[tm][tn][r];
            }
        }
    #pragma unroll
    for (int o = 16; o; o >>= 1) part += __shfl_xor(part, o, 32);
    if (lane == 0) atomicAdd(out, part);
}

// -------------------------------------------------------------------------
// launch
// -------------------------------------------------------------------------
extern "C" void kernel_launch(void* const* d_in, const int* in_sizes, int n_in,
                              void* d_out, int out_size, void* d_ws, size_t ws_size,
                              hipStream_t stream) {
    const float* phi = (const float*)d_in[0];   // (64, 512)
    const float* psi = (const float*)d_in[1];   // (64, 1024, 512)
    float* out = (float*)d_out;                 // scalar f32

    float* wsf   = (float*)d_ws;
    float* s_psi = wsf;                          // 64*1024 floats   (256 KB)
    float* gmat  = wsf + M_DIM * N_DIM;          // 64*64 floats     (16 KB)
    float* Smat  = gmat + M_DIM * M_DIM;         // 64*512*512 floats (64 MB)

    // normalization scales for psi (one wave per vector)
    k_psi_invnorm2<<<(M_DIM * N_DIM) / 8, 256, 0, stream>>>(psi, s_psi);

    // squared normalized gram of phi
    k_gram<<<dim3(M_DIM, M_DIM), 32, 0, stream>>>(phi, gmat);

    // loss accumulator = -m*n
    k_init<<<1, 1, 0, stream>>>(out);

    // S_i = psi_i^T diag(s) psi_i, upper-triangle blocks + mirror
    k_S<<<M_DIM * 10, 256, 0, stream>>>(psi, s_psi, Smat);

    // loss += sum_ij g_ij <S_i, S_j>  (WMMA over flattened S, fused reduce)
    k_b<<<256, 256, 0, stream>>>(Smat, gmat, out);
}